// iterative_reco_model_6588479832707
// MI455X (gfx1250) — compile-verified
//
#include <hip/hip_runtime.h>

#define N_ANGLES 720
#define N_DET    725
#define N_SAMP   768
#define IMG_W    512
#define IMG_H    512
#define N_GROUPS 46   // ceil(725/16)

typedef float v2f __attribute__((ext_vector_type(2)));
typedef float v8f __attribute__((ext_vector_type(8)));

// ---------------------------------------------------------------------------
// Kernel 1: upd = x + reco, staged into the reco slot of d_out (reused as the
// projection input; fixed up afterwards by Kernel 3).
// ---------------------------------------------------------------------------
__global__ void add_kernel(const float* __restrict__ x,
                           const float* __restrict__ reco,
                           float* __restrict__ upd, int n) {
  int i = blockIdx.x * blockDim.x + threadIdx.x;
  if (i < n) upd[i] = x[i] + reco[i];
}

// Kernel 3: restore the pass-through output reco.
__global__ void copy_kernel(const float* __restrict__ reco,
                            float* __restrict__ out, int n) {
  int i = blockIdx.x * blockDim.x + threadIdx.x;
  if (i < n) out[i] = reco[i];
}

// ---------------------------------------------------------------------------
// Bilinear sample with zero-outside semantics (per-tap validity, clamped
// addresses so no OOB loads, v_cndmask selects).
// ---------------------------------------------------------------------------
__device__ __forceinline__ float bilin(const float* __restrict__ img,
                                       float xs, float ys) {
  float xf = floorf(xs);
  float yf = floorf(ys);
  float wx = xs - xf;
  float wy = ys - yf;
  int ix = (int)xf, iy = (int)yf;
  int ix1 = ix + 1, iy1 = iy + 1;

  bool fx0 = (unsigned)ix  < (unsigned)IMG_W;
  bool fx1 = (unsigned)ix1 < (unsigned)IMG_W;
  bool fy0 = (unsigned)iy  < (unsigned)IMG_H;
  bool fy1 = (unsigned)iy1 < (unsigned)IMG_H;

  int xc0 = min(max(ix, 0), IMG_W - 1);
  int xc1 = min(max(ix1, 0), IMG_W - 1);
  int yc0 = min(max(iy, 0), IMG_H - 1);
  int yc1 = min(max(iy1, 0), IMG_H - 1);

  const float* r0 = img + yc0 * IMG_W;
  const float* r1 = img + yc1 * IMG_W;
  float l00 = r0[xc0];
  float l01 = r0[xc1];
  float l10 = r1[xc0];
  float l11 = r1[xc1];

  float v00 = (fx0 & fy0) ? l00 : 0.0f;
  float v01 = (fx1 & fy0) ? l01 : 0.0f;
  float v10 = (fx0 & fy1) ? l10 : 0.0f;
  float v11 = (fx1 & fy1) ? l11 : 0.0f;

  float top = fmaf(wx, v01 - v00, v00);
  float bot = fmaf(wx, v11 - v10, v10);
  return fmaf(wy, bot - top, top);
}

// ---------------------------------------------------------------------------
// Kernel 2: forward projection. One wave = one (angle, 16-detector group).
// Lane (m = lane&15, h = lane>>4) fills A[16x4] rows m, K = 2h, 2h+1.
// D = A * ones(4x16) + C accumulates per-detector row sums on the XDL pipe
// (v_wmma_f32_16x16x4_f32, exact f32). D column 0: lane 0 has M=0..7 in
// c[0..7], lane 16 has M=8..15.
// ---------------------------------------------------------------------------
__global__ __launch_bounds__(256)
void project_kernel(const float* __restrict__ img, float* __restrict__ sino) {
  const int angle = blockIdx.x;
  const int wave  = threadIdx.x >> 5;
  const int lane  = threadIdx.x & 31;
  const int group = blockIdx.y * 8 + wave;
  if (group >= N_GROUPS) return;          // wave-uniform exit

  const int m = lane & 15;
  const int h = lane >> 4;

  const float th = (float)angle * (3.14159265358979323846f / (float)N_ANGLES);
  float sn, c;
  sincosf(th, &sn, &c);

  const float cx = (IMG_W - 1) * 0.5f;    // 255.5
  const float cy = (IMG_H - 1) * 0.5f;
  const float det_c = (N_DET - 1) * 0.5f; // 362.0
  const float smp_c = (N_SAMP - 1) * 0.5f;// 383.5

  const float s  = (float)(group * 16 + m) - det_c;
  const float sc = s * c;                 // s*cos  (x pre-term)
  const float ss = s * sn;                // s*sin  (y pre-term)

  // wave-uniform min |s| over this 16-det group, for the dead-sample test
  const float s0 = (float)(group * 16) - det_c;
  const float s1 = s0 + 15.0f;
  const float smin  = (s0 > 0.0f) ? s0 : ((s1 < 0.0f) ? -s1 : 0.0f);
  const float smin2 = smin * smin;
  // sample contributes 0 when s^2 + t^2 > 2*(256.5)^2 = 131584.5
  const float R2SKIP = 131585.0f;

  v8f acc = {};
  v2f bones;
  bones.x = 1.0f;
  bones.y = 1.0f;
  const float hof = (float)(2 * h);

  for (int step = 0; step < N_SAMP / 4; ++step) {
    const float t0 = (float)(4 * step) - smp_c;
    const float t3 = t0 + 3.0f;
    const float tmin = (t0 > 0.0f) ? t0 : ((t3 < 0.0f) ? -t3 : 0.0f);
    if (fmaf(tmin, tmin, smin2) > R2SKIP) continue;   // uniform skip

    const float tA = t0 + hof;
    const float tB = tA + 1.0f;
    v2f a;
    a.x = bilin(img, fmaf(-tA, sn, sc) + cx, fmaf(tA, c, ss) + cy);
    a.y = bilin(img, fmaf(-tB, sn, sc) + cx, fmaf(tB, c, ss) + cy);

    // 8-arg form: (neg_a, A, neg_b, B, c_mod, C, reuse_a, reuse_b)
    acc = __builtin_amdgcn_wmma_f32_16x16x4_f32(
        false, a, false, bones, (short)0, acc, false, false);
  }

  if (lane == 0 || lane == 16) {
    const int dbase = group * 16 + (lane >> 1);  // +0 for lane 0, +8 for lane 16
    float* row = sino + (size_t)angle * N_DET;
#pragma unroll
    for (int i = 0; i < 8; ++i) {
      const int d = dbase + i;
      if (d < N_DET) row[d] = acc[i];            // * SAMP_SP (== 1.0)
    }
  }
}

// ---------------------------------------------------------------------------
extern "C" void kernel_launch(void* const* d_in, const int* in_sizes, int n_in,
                              void* d_out, int out_size, void* d_ws, size_t ws_size,
                              hipStream_t stream) {
  (void)in_sizes; (void)n_in; (void)out_size; (void)d_ws; (void)ws_size;
  const float* x    = (const float*)d_in[0];
  const float* reco = (const float*)d_in[1];

  float* out      = (float*)d_out;
  float* sino     = out;                                 // [720*725]
  float* out_reco = out + (size_t)N_ANGLES * N_DET;      // [512*512]

  const int n = IMG_W * IMG_H;

  // 1) stage upd = x + reco in the reco output slot (projection input)
  add_kernel<<<(n + 255) / 256, 256, 0, stream>>>(x, reco, out_reco, n);

  // 2) forward-project upd -> sino
  dim3 grid(N_ANGLES, (N_GROUPS + 7) / 8);
  project_kernel<<<grid, 256, 0, stream>>>(out_reco, sino);

  // 3) restore pass-through output: out_reco = reco
  copy_kernel<<<(n + 255) / 256, 256, 0, stream>>>(reco, out_reco, n);
}